// MovingAvgNorm_27608049778843
// MI455X (gfx1250) — compile-verified
//
#include <hip/hip_runtime.h>
#include <hip/hip_bf16.h>
#include <math.h>

// Problem constants (match reference: x = (16, 16384, 80) f32, K = 100)
#define B_      16
#define T_      16384
#define D_      80
#define K_      100
#define HALF_L  (K_ / 2)            // 50 left reflect pad
#define TILE_T  64                  // outputs per pipeline stage
#define STRIP   256                 // outputs per block (divides 16384)
#define NTILE   (STRIP / TILE_T)    // 4 chained tiles
#define ROWS0   (TILE_T + K_ - 1)   // 163 rows for the first tile
#define ROWS_S  (STRIP + K_ - 1)    // 355 rows per strip (linear LDS buffer)
#define NT      320                 // 10 waves of 32; = 4 * D_
#define SUBS    4
#define CH      (TILE_T / SUBS)     // 16 outputs per thread per tile

typedef unsigned int u32x4 __attribute__((ext_vector_type(4)));
typedef int          i32x4 __attribute__((ext_vector_type(4)));
typedef int          i32x8 __attribute__((ext_vector_type(8)));

// Issue one TDM 2-D tile load: nrows x 80 f32, row stride 80, global -> LDS.
__device__ __forceinline__ void tdm_load_rows(const float* __restrict__ src,
                                              const float* lds_dst, int nrows) {
  const unsigned long long ga = (unsigned long long)(const void*)src;
  const unsigned lds_addr = (unsigned)(uintptr_t)lds_dst;  // low32 = LDS byte offset

  u32x4 g0;
  g0[0] = 1u;                                                   // count=1, user desc
  g0[1] = lds_addr;                                             // LDS address (bytes)
  g0[2] = (unsigned)(ga & 0xFFFFFFFFull);                       // global_addr[31:0]
  g0[3] = (unsigned)((ga >> 32) & 0x01FFFFFFull) | (2u << 30);  // addr[56:32] | type=2

  i32x8 g1;
  g1[0] = (int)(2u << 16);                     // workgroup_mask=0, data_size=2 (4B)
  g1[1] = (int)((unsigned)D_ << 16);           // tensor_dim0 lo16 = 80
  g1[2] = (int)((unsigned)nrows << 16);        // tensor_dim0 hi16=0 | tensor_dim1 lo16
  g1[3] = (int)((unsigned)D_ << 16);           // tensor_dim1 hi16=0 | tile_dim0 = 80
  g1[4] = (int)((unsigned)nrows & 0xFFFFu);    // tile_dim1 = nrows | tile_dim2 = 0
  g1[5] = D_;                                  // tensor_dim0_stride lo32 = 80 elements
  g1[6] = 0;
  g1[7] = 0;

  const i32x4 gz  = {0, 0, 0, 0};
  const i32x8 gz8 = {0, 0, 0, 0, 0, 0, 0, 0};
  __builtin_amdgcn_tensor_load_to_lds(g0, g1, gz, gz, gz8, 0);
}

__device__ __forceinline__ int chunk_lo(int c) { return (c == 0) ? 0 : (ROWS0 + (c - 1) * TILE_T); }
__device__ __forceinline__ int chunk_hi(int c) { return ROWS0 + c * TILE_T; }

__global__ __launch_bounds__(NT)
void MovingAvgNorm_kernel(const float* __restrict__ x, float* __restrict__ out) {
  __shared__ float tile[ROWS_S * D_];   // 355*80*4 = 113,600 B (< 320 KB WGP LDS)

  const int b   = blockIdx.y;
  const int s0  = blockIdx.x * STRIP;   // first output t of this strip
  const int tid = threadIdx.x;
  const int gs  = s0 - HALF_L;          // global t of LDS row 0 (may be < 0)

  // Strip-level valid row range (reflection needed outside it; extreme strips only).
  const int r_lo = (gs < 0) ? -gs : 0;
  const int r_hi = (gs + ROWS_S > T_) ? (T_ - gs) : ROWS_S;

  const float* __restrict__ xb = x + (size_t)b * T_ * D_;
  const bool issuer = (tid < 32);       // wave 0 drives the TDM pipeline

  if (issuer) {
    // Prologue: chunk 0 (tile 0's rows) + chunk 1 prefetch.
    #pragma unroll
    for (int c = 0; c < 2; ++c) {
      const int a = max(chunk_lo(c), r_lo);
      const int e = min(chunk_hi(c), r_hi);
      tdm_load_rows(xb + (size_t)(gs + a) * D_, &tile[a * D_], e - a);
    }
    __builtin_amdgcn_s_wait_tensorcnt(1);   // chunk 0 complete; chunk 1 in flight
  }

  // Reflect-pad fixup rows (TDM never writes these; done once, up front).
  const int nfixLo = r_lo;
  const int nfixHi = ROWS_S - r_hi;
  const int nfix   = nfixLo + nfixHi;
  if (nfix) {
    for (int idx = tid; idx < nfix * D_; idx += NT) {
      int rr = idx / D_;
      int c  = idx - rr * D_;
      int r  = (rr < nfixLo) ? rr : (r_hi + (rr - nfixLo));
      int g  = gs + r;
      g = (g < 0) ? -g : ((g >= T_) ? (2 * T_ - 2 - g) : g);
      tile[r * D_ + c] = xb[(size_t)g * D_ + c];
    }
  }
  __syncthreads();   // tile 0 rows + all fixups visible

  const int d   = tid % D_;      // feature column
  const int sub = tid / D_;      // 0..3 sub-chunk within each tile
  const float invK  = 1.0f / (float)K_;
  const float invK1 = 1.0f / (float)(K_ - 1);

  for (int it = 0; it < NTILE; ++it) {
    // ---- Compute tile `it` while chunk it+1 streams in via TDM ----
    const int tl0 = it * TILE_T + sub * CH;   // row-space base of first window

    float s1 = 0.0f, s2 = 0.0f;
#pragma unroll 10
    for (int r = 0; r < K_; ++r) {
      float v = tile[(tl0 + r) * D_ + d];
      s1 += v;
      s2 = fmaf(v, v, s2);
    }

    for (int i = 0; i < CH; ++i) {
      const int o   = tl0 + i;                 // strip-local output / window base row
      const float m = s1 * invK;
      float var     = (s2 - (float)K_ * m * m) * invK1;
      var           = fmaxf(var, 0.0f);
      const float sd = sqrtf(var);
      const float xv = tile[(o + HALF_L) * D_ + d];
      out[((size_t)b * T_ + (size_t)(s0 + o)) * D_ + d] = (xv - m) / (sd + 1e-12f);
      if (i + 1 < CH) {
        const float nv = tile[(o + K_) * D_ + d];
        const float ov = tile[o * D_ + d];
        s1 += nv - ov;
        s2 += nv * nv - ov * ov;
      }
    }

    // ---- Advance the TDM pipeline ----
    if (it + 1 < NTILE) {
      if (issuer) {
        const int cnext = it + 2;
        if (cnext < NTILE) {   // prefetch chunk it+2 (rows strictly ahead of all readers)
          const int a = max(chunk_lo(cnext), r_lo);
          const int e = min(chunk_hi(cnext), r_hi);
          tdm_load_rows(xb + (size_t)(gs + a) * D_, &tile[a * D_], e - a);
          __builtin_amdgcn_s_wait_tensorcnt(1);   // chunk it+1 done; it+2 in flight
        } else {
          __builtin_amdgcn_s_wait_tensorcnt(0);   // last chunk done
        }
      }
      __syncthreads();   // publish chunk it+1 to all waves
    }
  }
}

extern "C" void kernel_launch(void* const* d_in, const int* in_sizes, int n_in,
                              void* d_out, int out_size, void* d_ws, size_t ws_size,
                              hipStream_t stream) {
  (void)in_sizes; (void)n_in; (void)out_size; (void)d_ws; (void)ws_size;
  const float* x = (const float*)d_in[0];
  float* out     = (float*)d_out;
  dim3 grid(T_ / STRIP, B_);   // (64, 16) = 1024 blocks
  dim3 block(NT);
  hipLaunchKernelGGL(MovingAvgNorm_kernel, grid, block, 0, stream, x, out);
}